// CrossModalityFusionGate_56427280334852
// MI455X (gfx1250) — compile-verified
//
#include <hip/hip_runtime.h>

// ---------------------------------------------------------------------------
// CrossModalityFusionGate fused kernels for gfx1250 (MI455X)
//   pass 1: h = W1*x + b1 via V_WMMA_F32_16X16X4_F32, reduce per-(b,group)
//           sum / sumsq into d_ws (atomics)
//   pass 2: recompute h, GroupNorm + LeakyReLU, logits = W2*hact via 4x
//           chained V_WMMA_F32_16X16X4_F32 (voxel-major output), tanh +
//           temperature softmax gate, blend with x, coalesced stores
// ---------------------------------------------------------------------------

typedef __attribute__((ext_vector_type(2))) float v2f;
typedef __attribute__((ext_vector_type(8))) float v8f;

#define DHW      2097152          // 128^3 voxels per (batch, channel)
#define Cc       4                // input/output channels
#define HID      16               // hidden channels
#define NGRP     4                // groupnorm groups
#define BATCH    2
#define GN_EPS   1e-5f
#define TEMP_EPS 1e-6f
#define SLOPE    0.1f
#define GROUP_N  8388608.0f       // (HID/NGRP) * DHW values per group

#define THREADS  256
#define WAVES    (THREADS / 32)
#define NB1      256              // blocks per batch, stats kernel
#define NB3      1024             // blocks per batch, main kernel
#define LDS_STR  20               // floats; 20*v + {0..3} covers 64 banks

// ------------------------------ pass 1 -------------------------------------
__global__ __launch_bounds__(THREADS)
void fg_stats(const float* __restrict__ x, const float* __restrict__ w1,
              const float* __restrict__ b1, float* __restrict__ ws)
{
    const int lane = threadIdx.x & 31;
    const int wv   = threadIdx.x >> 5;
    const int q    = lane & 15;        // A: voxel row / output: hidden channel
    const int hf   = lane >> 4;        // half-wave
    const int k0   = 2 * hf;           // K pair handled by this half-wave

    const int b       = blockIdx.x / NB1;
    const int blkInB  = blockIdx.x % NB1;
    const int waveInB = blkInB * WAVES + wv;
    const int wstride = NB1 * WAVES;                 // waves per batch
    const int tilesPerWave = (DHW / 16) / wstride;   // = 64

    // B matrix = W1^T : lane holds w1[n=q][k0], w1[q][k0+1]
    v2f bw;
    bw.x = w1[q * Cc + k0];
    bw.y = w1[q * Cc + k0 + 1];
    const float b1n = b1[q];

    const float* xb0 = x + (size_t)(b * Cc + k0) * DHW;
    const float* xb1 = x + (size_t)(b * Cc + k0 + 1) * DHW;

    float s = 0.f, sq = 0.f;
    for (int t = 0; t < tilesPerWave; ++t) {
        const int tb = (waveInB + t * wstride) * 16;
        v2f a;                               // A[m=q][k0], A[q][k0+1]
        a.x = xb0[tb + q];
        a.y = xb1[tb + q];
        v8f c = {};
        v8f h = __builtin_amdgcn_wmma_f32_16x16x4_f32(
                    false, a, false, bw, (short)0, c, false, false);
        #pragma unroll
        for (int r = 0; r < 8; ++r) {
            float hv = h[r] + b1n;           // pre-affine h (what GN averages)
            s  += hv;
            sq += hv * hv;
        }
    }
    // lanes with same hidden channel group -> one value per group
    s  += __shfl_xor(s, 16, 32);  sq += __shfl_xor(sq, 16, 32);
    s  += __shfl_xor(s,  1, 32);  sq += __shfl_xor(sq,  1, 32);
    s  += __shfl_xor(s,  2, 32);  sq += __shfl_xor(sq,  2, 32);

    __shared__ float part[WAVES][2][NGRP];
    if (lane < 16 && (q & 3) == 0) {
        part[wv][0][q >> 2] = s;
        part[wv][1][q >> 2] = sq;
    }
    __syncthreads();
    if (threadIdx.x < 8) {
        const int p = threadIdx.x >> 2;      // 0 = sum, 1 = sumsq
        const int g = threadIdx.x & 3;
        float acc = 0.f;
        #pragma unroll
        for (int w = 0; w < WAVES; ++w) acc += part[w][p][g];
        atomicAdd(&ws[b * 8 + p * 4 + g], acc);
    }
}

// ------------------------------ pass 2 -------------------------------------
__global__ __launch_bounds__(THREADS)
void fg_main(const float* __restrict__ x,  const float* __restrict__ w1,
             const float* __restrict__ b1, const float* __restrict__ gnw,
             const float* __restrict__ gnb, const float* __restrict__ w2,
             const float* __restrict__ bias2, const float* __restrict__ temp,
             const float* __restrict__ rscale, const float* __restrict__ ws,
             float* __restrict__ out)
{
    __shared__ float tr[WAVES][16 * LDS_STR];   // per-wave Hact tile

    const int lane = threadIdx.x & 31;
    const int wv   = threadIdx.x >> 5;
    const int q    = lane & 15;
    const int hf   = lane >> 4;
    const int k0   = 2 * hf;

    const int b       = blockIdx.x / NB3;
    const int blkInB  = blockIdx.x % NB3;
    const int waveInB = blkInB * WAVES + wv;
    const int wstride = NB3 * WAVES;
    const int tilesPerWave = (DHW / 16) / wstride;   // = 16

    v2f bw;
    bw.x = w1[q * Cc + k0];
    bw.y = w1[q * Cc + k0 + 1];
    const float b1n  = b1[q];
    const float gnwv = gnw[q];
    const float gnbv = gnb[q];

    // A2 = W2 zero-padded to 16 rows (M = output channel, K = hidden)
    v2f a2[4];
    #pragma unroll
    for (int j = 0; j < 4; ++j) {
        const int kk = 4 * j + k0;
        a2[j].x = (q < Cc) ? w2[q * HID + kk]     : 0.f;
        a2[j].y = (q < Cc) ? w2[q * HID + kk + 1] : 0.f;
    }
    const float bc0 = bias2[0], bc1 = bias2[1];
    const float bc2 = bias2[2], bc3 = bias2[3];
    const float tinv = 1.f / (fabsf(temp[0]) + TEMP_EPS);
    const float rs   = rscale[0];
    const float rs1  = 1.f - rs;

    // groupnorm stats for this lane's hidden channel
    const int   g    = q >> 2;
    const float mean = ws[b * 8 + g] * (1.f / GROUP_N);
    const float var  = ws[b * 8 + 4 + g] * (1.f / GROUP_N) - mean * mean;
    const float rstd = rsqrtf(var + GN_EPS);

    const float* xb0 = x + (size_t)(b * Cc + k0) * DHW;
    const float* xb1 = x + (size_t)(b * Cc + k0 + 1) * DHW;
    float* ob = out + (size_t)b * Cc * DHW;

    for (int t = 0; t < tilesPerWave; ++t) {
        const int tb = (waveInB + t * wstride) * 16;
        v2f a;
        a.x = xb0[tb + q];
        a.y = xb1[tb + q];

        // stage 1: h = x * W1^T  (M=voxel, N=hidden)
        v8f c0 = {};
        v8f h = __builtin_amdgcn_wmma_f32_16x16x4_f32(
                    false, a, false, bw, (short)0, c0, false, false);

        // normalize + affine + leaky relu; transpose into LDS as [voxel][hid]
        #pragma unroll
        for (int r = 0; r < 8; ++r) {
            float hv = h[r] + b1n;
            float y  = (hv - mean) * rstd * gnwv + gnbv;
            y = (y > 0.f) ? y : SLOPE * y;
            const int m = r + 8 * hf;              // voxel row of this element
            tr[wv][m * LDS_STR + q] = y;
        }
        asm volatile("s_wait_dscnt 0" ::: "memory");

        // stage 2: logits = W2pad x Hact^T  (M=channel, N=voxel), K=16 in 4 chunks
        v8f d2 = {};
        #pragma unroll
        for (int j = 0; j < 4; ++j) {
            v2f b2v = *(const v2f*)&tr[wv][q * LDS_STR + 4 * j + k0];
            d2 = __builtin_amdgcn_wmma_f32_16x16x4_f32(
                     false, a2[j], false, b2v, (short)0, d2, false, false);
        }

        // lanes 0-15 hold channels 0..3 (rows r=0..3) for voxel q
        float l0 = tanhf(d2[0] + bc0);
        float l1 = tanhf(d2[1] + bc1);
        float l2 = tanhf(d2[2] + bc2);
        float l3 = tanhf(d2[3] + bc3);
        float mx = fmaxf(fmaxf(l0, l1), fmaxf(l2, l3));
        float e0 = __expf((l0 - mx) * tinv);
        float e1 = __expf((l1 - mx) * tinv);
        float e2 = __expf((l2 - mx) * tinv);
        float e3 = __expf((l3 - mx) * tinv);
        float inv = 1.f / (e0 + e1 + e2 + e3);
        float gv0 = e0 * inv * rs + rs1;
        float gv1 = e1 * inv * rs + rs1;
        float gv2 = e2 * inv * rs + rs1;
        float gv3 = e3 * inv * rs + rs1;

        // hand channels 2/3 gates to the upper half-wave (which holds x ch2/3)
        float sg2 = __shfl_xor(gv2, 16, 32);
        float sg3 = __shfl_xor(gv3, 16, 32);
        float ga = hf ? sg2 : gv0;                 // gate for channel k0
        float gb = hf ? sg3 : gv1;                 // gate for channel k0+1

        ob[(size_t)k0 * DHW + tb + q]       = a.x * ga;
        ob[(size_t)(k0 + 1) * DHW + tb + q] = a.y * gb;
    }
}

// ------------------------------ launcher -----------------------------------
extern "C" void kernel_launch(void* const* d_in, const int* in_sizes, int n_in,
                              void* d_out, int out_size, void* d_ws, size_t ws_size,
                              hipStream_t stream) {
    const float* x     = (const float*)d_in[0];
    const float* w1    = (const float*)d_in[1];
    const float* b1    = (const float*)d_in[2];
    const float* gnw   = (const float*)d_in[3];
    const float* gnb   = (const float*)d_in[4];
    const float* w2    = (const float*)d_in[5];
    const float* bias2 = (const float*)d_in[6];
    const float* temp  = (const float*)d_in[7];
    const float* rsc   = (const float*)d_in[8];
    float* out = (float*)d_out;
    float* ws  = (float*)d_ws;

    (void)in_sizes; (void)n_in; (void)out_size; (void)ws_size;

    hipMemsetAsync(ws, 0, 16 * sizeof(float), stream);
    fg_stats<<<BATCH * NB1, THREADS, 0, stream>>>(x, w1, b1, ws);
    fg_main<<<BATCH * NB3, THREADS, 0, stream>>>(x, w1, b1, gnw, gnb, w2,
                                                 bias2, temp, rsc, ws, out);
}